// GMNet_1168231104975
// MI455X (gfx1250) — compile-verified
//
#include <hip/hip_runtime.h>

typedef float v2f __attribute__((ext_vector_type(2)));
typedef float v8f __attribute__((ext_vector_type(8)));

#define PSZ 512
#define BATCH 16
#define DDIM 128
#define TOT (PSZ * PSZ)

// ---------------------------------------------------------------------------
// Kernel 0: fused squared-norm terms  out[row] = 0.5*(||n_row||^2 + ||e_row||^2)
// rows are the flattened (b, p) index; 128 threads = one d-vector, coalesced.
// ---------------------------------------------------------------------------
__global__ void norms_kernel(const float* __restrict__ nA,
                             const float* __restrict__ eA,
                             float* __restrict__ out) {
  int row = blockIdx.x;
  int t = threadIdx.x;
  const float* np = nA + (size_t)row * DDIM;
  const float* ep = eA + (size_t)row * DDIM;
  float v = np[t] * np[t] + ep[t] * ep[t];
  __shared__ float red[DDIM];
  red[t] = v;
  __syncthreads();
  for (int s = DDIM / 2; s > 0; s >>= 1) {
    if (t < s) red[t] += red[t + s];
    __syncthreads();
  }
  if (t == 0) out[row] = 0.5f * red[0];
}

// ---------------------------------------------------------------------------
// Kernel 1: qm[b][j][i] = rn_nxt[b][j] + rn_cur[b][i]
//                         - (nnxt_j . ncur_i + enxt_j . ecur_i)
// One wave per 16x16 output tile; 64x V_WMMA_F32_16X16X4_F32 chained into
// one f32 accumulator (full f32 precision, matching the reference).
//
// ISA layouts (CDNA5 ISA):
//   A 16x4 f32 : lane l holds A[l&15][k0 + 2*(l>>4) + {0,1}] in {a.x, a.y}
//   B 4x16 f32 : lane l holds B[k0 + 2*(l>>4) + {0,1}][l&15] in {b.x, b.y}
//   D 16x16 f32: acc[v] = D[v + 8*(l>>4)][l&15]
// ---------------------------------------------------------------------------
__global__ __launch_bounds__(32) void qm_wmma_kernel(
    const float* __restrict__ ncur, const float* __restrict__ ecur,
    const float* __restrict__ nnxt, const float* __restrict__ enxt,
    const float* __restrict__ rncur, const float* __restrict__ rnnxt,
    float* __restrict__ qm_all) {
  const int it = blockIdx.x;   // i-tile (cols of qm, "m"/cur index)
  const int jt = blockIdx.y;   // j-tile (rows of qm, "n"/nxt index)
  const int b  = blockIdx.z;
  const int lane  = threadIdx.x;       // 0..31
  const int l15   = lane & 15;
  const int half  = lane >> 4;         // 0 or 1
  const int khalf = half * 2;          // K sub-offset 0 or 2

  // A rows come from the *next* embeddings (j index), B rows from *cur* (i).
  const float* An = nnxt + ((size_t)b * PSZ + jt * 16 + l15) * DDIM;
  const float* Ae = enxt + ((size_t)b * PSZ + jt * 16 + l15) * DDIM;
  const float* Bn = ncur + ((size_t)b * PSZ + it * 16 + l15) * DDIM;
  const float* Be = ecur + ((size_t)b * PSZ + it * 16 + l15) * DDIM;

  v8f acc = {};
#pragma unroll 4
  for (int k0 = 0; k0 < DDIM; k0 += 4) {
    v2f a0 = *(const v2f*)(An + k0 + khalf);
    v2f b0 = *(const v2f*)(Bn + k0 + khalf);
    acc = __builtin_amdgcn_wmma_f32_16x16x4_f32(false, a0, false, b0,
                                                (short)0, acc, false, false);
    v2f a1 = *(const v2f*)(Ae + k0 + khalf);
    v2f b1 = *(const v2f*)(Be + k0 + khalf);
    acc = __builtin_amdgcn_wmma_f32_16x16x4_f32(false, a1, false, b1,
                                                (short)0, acc, false, false);
  }

  const int i = it * 16 + l15;
  const float rc = rncur[b * PSZ + i];
  float* outb = qm_all + (size_t)b * TOT;
#pragma unroll
  for (int v = 0; v < 8; ++v) {
    int j = jt * 16 + v + 8 * half;
    outb[(size_t)j * PSZ + i] = rnnxt[b * PSZ + j] + rc - acc[v];
  }
}

// ---------------------------------------------------------------------------
// Kernel 2: 20-iteration projected-gradient QP, one 1024-thread WG per batch.
// X lives in d_out (global, L2-resident). Thread t owns half-row:
//   j = t>>1, cols [(t&1)*256, +256). Column pass: col i = t&511,
//   rows [(t>>9)*256, +256) -> coalesced 128B reads along rows.
// ---------------------------------------------------------------------------
__device__ inline float block_reduce_sum(float v, float* red, int tid) {
  red[tid] = v;
  __syncthreads();
  for (int s = 512; s > 0; s >>= 1) {
    if (tid < s) red[tid] += red[tid + s];
    __syncthreads();
  }
  float r = red[0];
  __syncthreads();
  return r;
}

__global__ __launch_bounds__(1024) void qp_kernel(
    const float* __restrict__ qm_all, float* __restrict__ x_all) {
  const int b = blockIdx.x;
  const float* qm = qm_all + (size_t)b * TOT;
  float* X = x_all + (size_t)b * TOT;
  const int tid = threadIdx.x;

  __shared__ float red[1024];
  __shared__ float part[1024];

  // ---- init: X = 1/512, s0 = sum(qm)/512, lr = 0.5/(sum(qm^2)+1e-8) ----
  const float c0 = 1.0f / (float)PSZ;
  float ps = 0.0f, psq = 0.0f;
  for (int idx = tid; idx < TOT; idx += 1024) {
    float qv = qm[idx];
    X[idx] = c0;
    ps += qv;
    psq += qv * qv;
  }
  float s = block_reduce_sum(ps, red, tid) * c0;
  float sq = block_reduce_sum(psq, red, tid);
  const float lr = 0.5f / (sq + 1e-8f);

  const int j = tid >> 1;
  const int cbase = (tid & 1) << 8;
  const size_t rowoff = (size_t)j * PSZ + cbase;
  const int ci = tid & 511;
  const int rbase = (tid >> 9) << 8;

  for (int iter = 0; iter < 20; ++iter) {
    const float coef = 2.0f * lr * s;

    // P2: gradient step + box clip, accumulate half-row sums
    float rp = 0.0f;
    for (int c = 0; c < 256; ++c) {
      size_t off = rowoff + c;
      float v = X[off] - coef * qm[off];
      v = fminf(fmaxf(v, 0.0f), 1.0f);
      X[off] = v;
      rp += v;
    }
    part[tid] = rp;
    __syncthreads();

    // P3: row normalize (own elements only; partner partial via LDS)
    const float rinv = 1.0f / (part[tid] + part[tid ^ 1] + 1e-8f);
    for (int c = 0; c < 256; ++c) X[rowoff + c] *= rinv;
    __threadfence_block();
    __syncthreads();

    // P3b: column partial sums (transposed, coalesced along rows)
    float cp = 0.0f;
    for (int r = 0; r < 256; ++r) cp += X[(size_t)(rbase + r) * PSZ + ci];
    part[tid] = cp;
    __syncthreads();

    // P4: column scale (col sum <= 2) fused with next iteration's s = qm.X
    float sp = 0.0f;
    for (int c = 0; c < 256; ++c) {
      int i = cbase + c;
      float cs = part[i] + part[i + 512];
      float f = fminf(1.0f, 2.0f / (cs + 1e-8f));
      size_t off = rowoff + c;
      float v = X[off] * f;
      X[off] = v;
      sp += qm[off] * v;
    }
    __threadfence_block();
    __syncthreads();
    s = block_reduce_sum(sp, red, tid);
  }
}

// ---------------------------------------------------------------------------
extern "C" void kernel_launch(void* const* d_in, const int* in_sizes, int n_in,
                              void* d_out, int out_size, void* d_ws,
                              size_t ws_size, hipStream_t stream) {
  const float* ncur = (const float*)d_in[0];
  const float* ecur = (const float*)d_in[1];
  const float* nnxt = (const float*)d_in[2];
  const float* enxt = (const float*)d_in[3];
  float* X = (float*)d_out;                       // (16, 512*512) f32
  float* qm = (float*)d_ws;                       // 16 MB
  float* rncur = qm + (size_t)BATCH * TOT;        // 32 KB
  float* rnnxt = rncur + BATCH * PSZ;             // 32 KB

  norms_kernel<<<BATCH * PSZ, DDIM, 0, stream>>>(ncur, ecur, rncur);
  norms_kernel<<<BATCH * PSZ, DDIM, 0, stream>>>(nnxt, enxt, rnnxt);

  dim3 gq(PSZ / 16, PSZ / 16, BATCH);             // 32 x 32 x 16 tiles
  qm_wmma_kernel<<<gq, 32, 0, stream>>>(ncur, ecur, nnxt, enxt, rncur, rnnxt,
                                        qm);

  qp_kernel<<<BATCH, 1024, 0, stream>>>(qm, X);
}